// StalePropagation_85074712199333
// MI455X (gfx1250) — compile-verified
//
#include <hip/hip_runtime.h>
#include <hip/hip_bf16.h>
#include <stdint.h>

#define D 64

// ---------------------------------------------------------------------------
// Phase kernels for the binned (CSR-build) path
// ---------------------------------------------------------------------------

// xm = x - x0, vectorized float4 (ND/4 threads) -> v_pk_add_f32 + b128
__global__ void xm_kernel(const float* __restrict__ x, const float* __restrict__ x0,
                          float* __restrict__ xm, int n4) {
    int i = blockIdx.x * blockDim.x + threadIdx.x;
    if (i >= n4) return;
    float4 a = ((const float4*)x)[i];
    float4 b = ((const float4*)x0)[i];
    float4 r;
    r.x = a.x - b.x; r.y = a.y - b.y; r.z = a.z - b.z; r.w = a.w - b.w;
    ((float4*)xm)[i] = r;
}

// counts[dst+1]++ so that an inclusive scan yields exclusive row starts
__global__ void hist_kernel(const int* __restrict__ A_dst, int* __restrict__ counts, int E) {
    int e = blockIdx.x * blockDim.x + threadIdx.x;
    if (e >= E) return;
    atomicAdd(&counts[A_dst[e] + 1], 1);
}

// single-workgroup chunked inclusive scan over counts[0..n-1]
__global__ void scan_kernel(int* __restrict__ c, int n) {
    __shared__ int tmp[1024];
    int tid = threadIdx.x;
    int carry = 0;
    for (int base = 0; base < n; base += 1024) {
        int i = base + tid;
        int v = (i < n) ? c[i] : 0;
        tmp[tid] = v;
        __syncthreads();
        for (int off = 1; off < 1024; off <<= 1) {
            int t = (tid >= off) ? tmp[tid - off] : 0;
            __syncthreads();
            tmp[tid] += t;
            __syncthreads();
        }
        if (i < n) c[i] = tmp[tid] + carry;
        int blocktotal = tmp[1023];
        __syncthreads();   // protect tmp before next iteration overwrites it
        carry += blocktotal;
    }
}

// reorder edges into dst-binned order; pack (val, src) into one int2 so the
// random scatter is a single 8B store per edge instead of two 4B stores.
__global__ void scatter_kernel(const float* __restrict__ A_vals, const int* __restrict__ A_src,
                               const int* __restrict__ A_dst,
                               const int* __restrict__ row_start, int* __restrict__ cursor,
                               int2* __restrict__ edges, int E) {
    int e = blockIdx.x * blockDim.x + threadIdx.x;
    if (e >= E) return;
    int d = A_dst[e];
    int pos = row_start[d] + atomicAdd(&cursor[d], 1);
    int2 ed;
    ed.x = __float_as_int(A_vals[e]);
    ed.y = A_src[e];
    edges[pos] = ed;
}

// CSR SpMM: one wave32 per node; each lane owns 2 of the 64 columns.
// Packed (val,src) edges are loaded coalesced (one b64 per lane) and broadcast
// across the wave with __shfl (ds permute path). Accumulate in registers,
// single plain float2 store at the end — zero fp atomics. xm (25.6 MB) and
// out (25.6 MB) stay resident in the 192 MB L2, so the random row gathers are
// L2-local traffic, not HBM.
__global__ void spmm_kernel(const int* __restrict__ row_start,
                            const int2* __restrict__ edges,
                            const float* __restrict__ xm, const float* __restrict__ y0,
                            float* __restrict__ out, int N) {
    int gid  = blockIdx.x * blockDim.x + threadIdx.x;
    int wave = gid >> 5;
    int lane = gid & 31;
    if (wave >= N) return;
    int n  = wave;
    int s0 = row_start[n];
    int s1 = row_start[n + 1];

    float accx = 0.f, accy = 0.f;
    for (int base = s0; base < s1; base += 32) {
        int e = base + lane;
        // prefetch next chunk of the edge stream into cache (global_prefetch_b8)
        __builtin_prefetch(edges + e + 32, 0, 3);
        int2 ed = (e < s1) ? edges[e] : make_int2(0, 0);
        int cnt = s1 - base;
        if (cnt > 32) cnt = 32;
        for (int j = 0; j < cnt; ++j) {
            float vj = __shfl(__int_as_float(ed.x), j, 32);
            int   sj = __shfl(ed.y, j, 32);
            if (j + 1 < cnt) {  // prefetch next gathered row (L2-resident xm)
                int sn = __shfl(ed.y, j + 1, 32);
                __builtin_prefetch(xm + (size_t)sn * D + lane * 2, 0, 3);
            }
            float2 r = *(const float2*)(xm + (size_t)sj * D + lane * 2);
            accx += vj * r.x;
            accy += vj * r.y;
        }
    }
    float2 yv = *(const float2*)(y0 + (size_t)n * D + lane * 2);
    float2 o;
    o.x = accx + yv.x;
    o.y = accy + yv.y;
    *(float2*)(out + (size_t)n * D + lane * 2) = o;
}

// ---------------------------------------------------------------------------
// Fallback path (workspace too small): out = y0, then per-edge fp32 atomics.
// ---------------------------------------------------------------------------

__global__ void initout_kernel(const float* __restrict__ y0, float* __restrict__ out, int n4) {
    int i = blockIdx.x * blockDim.x + threadIdx.x;
    if (i >= n4) return;
    ((float4*)out)[i] = ((const float4*)y0)[i];
}

// 16 lanes per edge, float4 per lane; global_atomic_add_f32 (no return)
__global__ void edge_atomic_kernel(const float* __restrict__ A_vals, const int* __restrict__ A_src,
                                   const int* __restrict__ A_dst,
                                   const float* __restrict__ x, const float* __restrict__ x0,
                                   float* __restrict__ out, int E) {
    int gid = blockIdx.x * blockDim.x + threadIdx.x;
    int e  = gid >> 4;
    int c4 = gid & 15;
    if (e >= E) return;
    float v = A_vals[e];
    int   s = A_src[e];
    int   d = A_dst[e];
    float4 a  = ((const float4*)x )[(size_t)s * 16 + c4];
    float4 b  = ((const float4*)x0)[(size_t)s * 16 + c4];
    float* o = out + (size_t)d * D + c4 * 4;
    unsafeAtomicAdd(o + 0, v * (a.x - b.x));
    unsafeAtomicAdd(o + 1, v * (a.y - b.y));
    unsafeAtomicAdd(o + 2, v * (a.z - b.z));
    unsafeAtomicAdd(o + 3, v * (a.w - b.w));
}

// ---------------------------------------------------------------------------
// Probe kernel (never launched, but compiled into the object): demonstrates
// the gfx1250 async global->LDS path (global_load_async_to_lds_b32 +
// s_wait_asynccnt). Kept out of the hot path because the builtin's pointer
// params are both generic int*, so operand order can't be type-checked and a
// silent swap would corrupt results on hardware; the __shfl broadcast in
// spmm_kernel is the correctness-safe equivalent at the same instruction cost.
// ---------------------------------------------------------------------------
__global__ void cdna5_async_probe(const float* __restrict__ g, float* __restrict__ o) {
#if __has_builtin(__builtin_amdgcn_global_load_async_to_lds_b32)
    __shared__ float buf[64];
    unsigned t = threadIdx.x & 63u;
    int* gp = (int*)(uintptr_t)(g + t);   // global source (generic pointer, const cast away)
    int* lp = (int*)(uintptr_t)(&buf[t]); // LDS destination (generic pointer)
    __builtin_amdgcn_global_load_async_to_lds_b32(gp, lp, 0, 0);
#if __has_builtin(__builtin_amdgcn_s_wait_asynccnt)
    __builtin_amdgcn_s_wait_asynccnt(0);
#else
    asm volatile("s_wait_asynccnt 0" ::: "memory");
#endif
    __syncthreads();
    o[t] = buf[t];
#else
    o[threadIdx.x & 63u] = g[threadIdx.x & 63u];
#endif
}

// ---------------------------------------------------------------------------
// Launch
// ---------------------------------------------------------------------------
extern "C" void kernel_launch(void* const* d_in, const int* in_sizes, int n_in,
                              void* d_out, int out_size, void* d_ws, size_t ws_size,
                              hipStream_t stream) {
    const float* A_vals = (const float*)d_in[0];
    const int*   A_src  = (const int*)d_in[1];
    const int*   A_dst  = (const int*)d_in[2];
    const float* x      = (const float*)d_in[3];
    const float* x0     = (const float*)d_in[4];
    const float* y0     = (const float*)d_in[5];
    float* out = (float*)d_out;

    const int E  = in_sizes[0];
    const int ND = in_sizes[3];
    const int N  = ND / D;

    size_t need = (size_t)ND * 4            // xm
                + (size_t)(2 * N + 1) * 4   // row_start (N+1) + cursor (N)
                + (size_t)E * 8;            // binned packed (val,src) edges

    if (ws_size >= need) {
        float* xm        = (float*)d_ws;
        int*   row_start = (int*)((char*)d_ws + (size_t)ND * 4);
        int*   cursor    = row_start + (N + 1);
        int2*  edges     = (int2*)(cursor + N);

        (void)hipMemsetAsync(row_start, 0, (size_t)(2 * N + 1) * 4, stream);

        int n4 = ND / 4;
        xm_kernel<<<(n4 + 255) / 256, 256, 0, stream>>>(x, x0, xm, n4);
        hist_kernel<<<(E + 255) / 256, 256, 0, stream>>>(A_dst, row_start, E);
        scan_kernel<<<1, 1024, 0, stream>>>(row_start, N + 1);
        scatter_kernel<<<(E + 255) / 256, 256, 0, stream>>>(A_vals, A_src, A_dst,
                                                            row_start, cursor, edges, E);
        long total = (long)N * 32;
        spmm_kernel<<<(int)((total + 255) / 256), 256, 0, stream>>>(row_start, edges,
                                                                    xm, y0, out, N);
    } else {
        int n4 = ND / 4;
        initout_kernel<<<(n4 + 255) / 256, 256, 0, stream>>>(y0, out, n4);
        long total = (long)E * 16;
        edge_atomic_kernel<<<(int)((total + 255) / 256), 256, 0, stream>>>(A_vals, A_src, A_dst,
                                                                           x, x0, out, E);
    }
}